// MultiHeadAttention_KVCache_2113123909937
// MI455X (gfx1250) — compile-verified
//
#include <hip/hip_runtime.h>
#include <hip/hip_bf16.h>

// MHA decode step with KV cache, MI455X (gfx1250, wave32).
// out = softmax(q·K/sqrt(128))·V projected; k/v = concat(past, new).
// Bandwidth-bound (~2.4 GB traffic); GEMMs use native fp32 WMMA 16x16x4.
// X is zero-padded to 16 rows in workspace so the WMMA A-operand load is
// branchless (no exec-mask churn in the inner loop).

#define Bsz   8
#define Hn    32
#define Dh    128
#define PASTL 4095
#define Lkv   4096
#define DM    4096   // d_model
#define Nn    4096   // H*Dh
#define NCHUNK 8
#define CHUNK  512   // Lkv / NCHUNK

typedef float v2f __attribute__((ext_vector_type(2)));
typedef float v8f __attribute__((ext_vector_type(8)));

// -------- pad X (8 x 4096) -> Xp (16 x 4096), rows 8..15 zeroed --------------
__global__ __launch_bounds__(256) void pad16(
    const float* __restrict__ src, float* __restrict__ dst)
{
  const int i = blockIdx.x * 256 + threadIdx.x;      // over 16*DM/4 float4s
  if (i < 16 * DM / 4) {
    const int row = i / (DM / 4);
    float4 val = make_float4(0.f, 0.f, 0.f, 0.f);
    if (row < Bsz) val = ((const float4*)src)[i];
    ((float4*)dst)[i] = val;
  }
}

// -------- zero rows 8..15 of a padded 16 x 4096 buffer -----------------------
__global__ __launch_bounds__(256) void zero_tail16(float* __restrict__ dst)
{
  const int i = blockIdx.x * 256 + threadIdx.x;      // over 8*DM/4 float4s
  if (i < Bsz * DM / 4)
    ((float4*)(dst))[Bsz * DM / 4 + i] = make_float4(0.f, 0.f, 0.f, 0.f);
}

// -------- Y(8 x 4096) = Xp(16 x 4096) @ W(4096 x 4096) + bias, fp32 WMMA -----
// One 16-column tile per block; 4 waves split K; LDS reduce (deterministic).
// A 16x4 f32 layout: lanes0-15 M=0..15 {v0:K=0, v1:K=1}; lanes16-31 {K=2,K=3}.
// B 4x16 f32 layout: v0 lanes0-15 row K=0, lanes16-31 row K=2; v1: K=1/K=3.
// D 16x16 f32: VGPR r -> row r (lanes0-15), row r+8 (lanes16-31).
__global__ __launch_bounds__(128) void gemm8_wmma(
    const float* __restrict__ Xp, const float* __restrict__ W,
    const float* __restrict__ bias, float* __restrict__ Y)
{
  __shared__ float red[4][32][8];
  const int lane = threadIdx.x & 31;
  const int wave = threadIdx.x >> 5;
  const int half = lane >> 4;      // 0: lanes 0-15, 1: lanes 16-31
  const int m    = lane & 15;
  const int col  = blockIdx.x * 16 + m;

  v8f acc = {};
  const int kBeg = wave * (DM / 4);   // 1024 K per wave
  const int kEnd = kBeg + (DM / 4);
  #pragma unroll 4
  for (int k0 = kBeg; k0 < kEnd; k0 += 4) {
    const int ka = k0 + 2 * half;
    const v2f a = *(const v2f*)(Xp + (size_t)m * DM + ka);  // padded: no guard
    v2f b;
    b.x = W[(size_t)ka * Nn + col];
    b.y = W[(size_t)(ka + 1) * Nn + col];
    acc = __builtin_amdgcn_wmma_f32_16x16x4_f32(
        /*neg_a=*/false, a, /*neg_b=*/false, b,
        /*c_mod=*/(short)0, acc, /*reuse_a=*/false, /*reuse_b=*/false);
  }
  #pragma unroll
  for (int r = 0; r < 8; ++r) red[wave][lane][r] = acc[r];
  __syncthreads();
  if (wave == 0) {
    #pragma unroll
    for (int r = 0; r < 8; ++r)
      acc[r] = red[0][lane][r] + red[1][lane][r] + red[2][lane][r] + red[3][lane][r];
    if (half == 0) {                 // rows 0..7 of D (== the 8 batch rows)
      const float bb = bias[col];
      #pragma unroll
      for (int r = 0; r < 8; ++r)
        Y[(size_t)r * Nn + col] = acc[r] + bb;
    }
  }
}

// -------- Fused cache-copy + flash-decoding partial softmax ------------------
// grid (NCHUNK, B*H), 256 threads. Each block: 512 keys of one (b,h).
__global__ __launch_bounds__(256) void attn_chunk(
    const float* __restrict__ q, const float* __restrict__ k_new,
    const float* __restrict__ v_new,
    const float* __restrict__ past_k, const float* __restrict__ past_v,
    float* __restrict__ k_out, float* __restrict__ v_out,
    float* __restrict__ part)
{
  const int c  = blockIdx.x;
  const int bh = blockIdx.y;
  const int b  = bh >> 5;
  const int h  = bh & 31;
  const int t0 = c * CHUNK;
  const int tid = threadIdx.x;

  __shared__ float p[CHUNK];
  __shared__ float sred[256];

  // --- copy this chunk of K and V into the output caches (float4 coalesced)
  {
    const size_t dstBase = ((size_t)bh * Lkv + t0) * Dh;
    for (int i = tid; i < CHUNK * (Dh / 4); i += 256) {
      const int  tr = i >> 5;               // row within chunk
      const int  t  = t0 + tr;
      const int  d4 = (i & 31) * 4;
      const float* ks = (t < PASTL) ? past_k + ((size_t)bh * PASTL + t) * Dh + d4
                                    : k_new + (size_t)b * Nn + h * Dh + d4;
      const float* vs = (t < PASTL) ? past_v + ((size_t)bh * PASTL + t) * Dh + d4
                                    : v_new + (size_t)b * Nn + h * Dh + d4;
      *(float4*)(k_out + dstBase + (size_t)tr * Dh + d4) = *(const float4*)ks;
      *(float4*)(v_out + dstBase + (size_t)tr * Dh + d4) = *(const float4*)vs;
    }
  }

  // --- scores: 2 keys per thread (keys t0+tid and t0+256+tid)
  const float scale = 0.08838834764831844f;  // 1/sqrt(128)
  const float* qv = q + (size_t)b * Nn + h * Dh;
  float s[2];
  #pragma unroll
  for (int j = 0; j < 2; ++j) {
    const int t = t0 + tid + j * 256;
    const float* krow = (t < PASTL) ? past_k + ((size_t)bh * PASTL + t) * Dh
                                    : k_new + (size_t)b * Nn + h * Dh;
    float dot = 0.f;
    #pragma unroll 8
    for (int d = 0; d < Dh; d += 4) {
      const float4 kk = *(const float4*)(krow + d);
      const float4 qq = *(const float4*)(qv + d);
      dot += kk.x * qq.x + kk.y * qq.y + kk.z * qq.z + kk.w * qq.w;
    }
    s[j] = dot * scale;
  }

  // --- block max
  sred[tid] = fmaxf(s[0], s[1]);
  __syncthreads();
  for (int off = 128; off > 0; off >>= 1) {
    if (tid < off) sred[tid] = fmaxf(sred[tid], sred[tid + off]);
    __syncthreads();
  }
  const float mc = sred[0];
  __syncthreads();

  // --- exp + block sum
  const float p0 = __expf(s[0] - mc), p1 = __expf(s[1] - mc);
  p[tid] = p0; p[tid + 256] = p1;
  sred[tid] = p0 + p1;
  __syncthreads();
  for (int off = 128; off > 0; off >>= 1) {
    if (tid < off) sred[tid] += sred[tid + off];
    __syncthreads();
  }
  const float lc = sred[0];
  __syncthreads();

  // --- acc[d] = sum_t p[t] * V[t][d] ; lane-per-d -> coalesced V reads
  const int d   = tid & 127;
  const int hlf = tid >> 7;                 // two halves of 256 keys each
  float accv = 0.f;
  for (int tt = 0; tt < 256; ++tt) {
    const int ti = hlf * 256 + tt;
    const int t  = t0 + ti;
    const float* vrow = (t < PASTL) ? past_v + ((size_t)bh * PASTL + t) * Dh
                                    : v_new + (size_t)b * Nn + h * Dh;
    accv += p[ti] * vrow[d];
  }
  __syncthreads();
  sred[tid] = accv;
  __syncthreads();
  if (tid < 128) {
    float* pp = part + ((size_t)bh * NCHUNK + c) * (Dh + 2);
    pp[2 + tid] = sred[tid] + sred[tid + 128];
    if (tid == 0) { pp[0] = mc; pp[1] = lc; }
  }
}

// -------- merge NCHUNK partials -> padded context (16 x 4096, rows 0-7) ------
__global__ __launch_bounds__(128) void attn_combine(
    const float* __restrict__ part, float* __restrict__ ctxp)
{
  const int bh = blockIdx.x;
  const int d  = threadIdx.x;
  const float* pp = part + (size_t)bh * NCHUNK * (Dh + 2);
  float M = -INFINITY;
  #pragma unroll
  for (int c = 0; c < NCHUNK; ++c) M = fmaxf(M, pp[c * (Dh + 2)]);
  float L = 0.f, A = 0.f;
  #pragma unroll
  for (int c = 0; c < NCHUNK; ++c) {
    const float w = __expf(pp[c * (Dh + 2)] - M);
    L += w * pp[c * (Dh + 2) + 1];
    A += w * pp[c * (Dh + 2) + 2 + d];
  }
  const int b = bh >> 5, h = bh & 31;
  ctxp[(size_t)b * Nn + h * Dh + d] = A / L;
}

extern "C" void kernel_launch(void* const* d_in, const int* in_sizes, int n_in,
                              void* d_out, int out_size, void* d_ws, size_t ws_size,
                              hipStream_t stream) {
  const float* x      = (const float*)d_in[0];
  const float* past_k = (const float*)d_in[1];
  const float* past_v = (const float*)d_in[2];
  const float* Wq = (const float*)d_in[3]; const float* bq = (const float*)d_in[4];
  const float* Wk = (const float*)d_in[5]; const float* bk = (const float*)d_in[6];
  const float* Wv = (const float*)d_in[7]; const float* bv = (const float*)d_in[8];
  const float* Wo = (const float*)d_in[9]; const float* bo = (const float*)d_in[10];
  // d_in[11] = max_seqlen (4096); concat length == 4096 so no truncation occurs.

  float* out   = (float*)d_out;                        // (8, 4096)
  float* k_out = out + (size_t)Bsz * DM;               // (8,32,4096,128)
  float* v_out = k_out + (size_t)Bsz * Hn * Lkv * Dh;  // (8,32,4096,128)

  // workspace layout (floats), ~1.94 MB total
  float* ws     = (float*)d_ws;
  float* xpad   = ws;                  // 16*4096 = 65536
  float* ctxpad = ws + 65536;          // 16*4096 = 65536
  float* qb     = ws + 131072;         // 32768
  float* k_new  = ws + 163840;         // 32768
  float* v_new  = ws + 196608;         // 32768
  float* part   = ws + 229376;         // B*H*NCHUNK*(Dh+2) = 266240

  pad16     <<<dim3(64), dim3(256), 0, stream>>>(x, xpad);
  zero_tail16<<<dim3(32), dim3(256), 0, stream>>>(ctxpad);

  gemm8_wmma<<<dim3(Nn / 16), dim3(128), 0, stream>>>(xpad, Wq, bq, qb);
  gemm8_wmma<<<dim3(Nn / 16), dim3(128), 0, stream>>>(xpad, Wk, bk, k_new);
  gemm8_wmma<<<dim3(Nn / 16), dim3(128), 0, stream>>>(xpad, Wv, bv, v_new);

  attn_chunk<<<dim3(NCHUNK, Bsz * Hn), dim3(256), 0, stream>>>(
      qb, k_new, v_new, past_k, past_v, k_out, v_out, part);
  attn_combine<<<dim3(Bsz * Hn), dim3(128), 0, stream>>>(part, ctxpad);

  gemm8_wmma<<<dim3(Nn / 16), dim3(128), 0, stream>>>(ctxpad, Wo, bo, out);
}